// SamplingLoss_87153476370783
// MI455X (gfx1250) — compile-verified
//
#include <hip/hip_runtime.h>

// CDNA5 / gfx1250, wave32. Fused sampled-softmax loss.
// Precision-preserving path: V_WMMA_F32_16X16X4_F32 on fp32 inputs.

typedef float v2f __attribute__((ext_vector_type(2)));
typedef float v8f __attribute__((ext_vector_type(8)));

#define H      128     // hidden dim
#define KSMP   256     // number of noise samples
#define MTILE  16      // rows per workgroup
#define HPAD   130     // padded LDS row stride (even -> 8B aligned float2, breaks bank conflicts)

// ---------------------------------------------------------------------------
// Kernel A: gather sampled weight rows into workspace + fused bias adjустment
//   Ws[k][h]  = weight[samples[k]][h]
//   badj[k]   = bias[samples[k]] - noise_logits[samples[k]]
// grid = KSMP blocks x H threads (coalesced row copy)
// ---------------------------------------------------------------------------
__global__ void gather_kernel(const float* __restrict__ weight,
                              const float* __restrict__ bias,
                              const float* __restrict__ noise_logits,
                              const int*   __restrict__ samples,
                              float* __restrict__ Ws,
                              float* __restrict__ badj) {
    const int k = blockIdx.x;
    const int s = samples[k];
    Ws[k * H + threadIdx.x] = weight[(long long)s * H + threadIdx.x];
    if (threadIdx.x == 0) badj[k] = bias[s] - noise_logits[s];
}

// ---------------------------------------------------------------------------
// Kernel B: fused scores + stable logsumexp (G = 2) per row, one partial/WG.
// 256 threads = 8 waves. Wave w computes column tiles 2w and 2w+1 with
// v_wmma_f32_16x16x4_f32, accumulating over H=128 in 32 K-steps.
//
// WMMA fp32 operand layouts (ISA 7.12.2):
//   A 16x4 : lane m = lane&15; VGPR0/1 hold K = 2*(lane>>4) + {0,1}
//   B 4x16 : lane n = lane&15; VGPR0/1 hold K = 2*(lane>>4) + {0,1}
//   C 16x16: VGPR r -> row (r + 8*(lane>>4)), col = lane&15
// ---------------------------------------------------------------------------
__global__ __launch_bounds__(256) void score_loss_kernel(
    const float* __restrict__ weight,
    const float* __restrict__ bias,
    const float* __restrict__ hiddens,
    const float* __restrict__ noise_logits,
    const int*   __restrict__ targets,
    const float* __restrict__ Ws,
    const float* __restrict__ badj,
    float* __restrict__ partials) {

    __shared__ float shA[MTILE * HPAD];   // hiddens tile, padded
    __shared__ float sh_o0[MTILE];        // target-term output[:,0] per row
    __shared__ float sh_pm[8][MTILE];     // per-wave running max per row
    __shared__ float sh_ps[8][MTILE];     // per-wave sum(exp(2*(v-max))) per row
    __shared__ float sh_row[MTILE];       // per-row  (lse - o0)

    const int tid  = threadIdx.x;
    const int row0 = blockIdx.x * MTILE;

    // ---- phase 0: stage 16x128 hiddens tile into LDS (coalesced) ----------
    for (int idx = tid; idx < MTILE * H; idx += 256) {
        const int r = idx >> 7, c = idx & 127;
        shA[r * HPAD + c] = hiddens[row0 * H + idx];
    }
    __syncthreads();

    const int w    = tid >> 5;
    const int lane = tid & 31;

    // ---- phase 1: target scores (wave w -> rows 2w, 2w+1) ------------------
    // coalesced gathered reads of weight[targets[row]] (16.8 MB total stream)
    for (int rr = 0; rr < 2; ++rr) {
        const int r = 2 * w + rr;
        const int t = targets[row0 + r];
        const float* wt = weight + (long long)t * H;
        float s = 0.f;
        #pragma unroll
        for (int j = 0; j < 4; ++j)
            s += shA[r * HPAD + lane + 32 * j] * wt[lane + 32 * j];
        #pragma unroll
        for (int m = 16; m >= 1; m >>= 1) s += __shfl_xor(s, m, 32);
        if (lane == 0) sh_o0[r] = s + bias[t] - noise_logits[t];
    }

    // ---- phase 2: WMMA GEMM, wave w owns 32 sample-columns ----------------
    const int half = lane >> 4;
    const int l16  = lane & 15;
    const int col0 = (2 * w)     * 16 + l16;
    const int col1 = (2 * w + 1) * 16 + l16;
    const float* ap  = shA + l16 * HPAD;       // A row for this lane
    const float* bp0 = Ws  + col0 * H;         // B rows (Ws is L2-resident)
    const float* bp1 = Ws  + col1 * H;

    v8f acc0 = {};
    v8f acc1 = {};
    #pragma unroll
    for (int kk = 0; kk < 32; ++kk) {
        const int kb = kk * 4 + 2 * half;
        const v2f a  = *(const v2f*)(ap  + kb);   // ds_load_b64
        const v2f b0 = *(const v2f*)(bp0 + kb);   // global_load_b64 (L2 hit)
        const v2f b1 = *(const v2f*)(bp1 + kb);
        acc0 = __builtin_amdgcn_wmma_f32_16x16x4_f32(
                   false, a, false, b0, (short)0, acc0, false, false);
        acc1 = __builtin_amdgcn_wmma_f32_16x16x4_f32(
                   false, a, false, b1, (short)0, acc1, false, false);
    }

    const float ba0 = badj[col0];
    const float ba1 = badj[col1];

    // ---- per-row (max, sum exp(2*(v-max))) over this wave's 32 columns ----
    #pragma unroll
    for (int r = 0; r < 8; ++r) {
        float v0 = acc0[r] + ba0;
        float v1 = acc1[r] + ba1;
        float mx = fmaxf(v0, v1);
        #pragma unroll
        for (int m = 8; m >= 1; m >>= 1) mx = fmaxf(mx, __shfl_xor(mx, m, 32));
        float se = __expf(2.f * (v0 - mx)) + __expf(2.f * (v1 - mx));
        #pragma unroll
        for (int m = 8; m >= 1; m >>= 1) se += __shfl_xor(se, m, 32);
        const int rowm = r + 8 * half;
        if (l16 == 0) { sh_pm[w][rowm] = mx; sh_ps[w][rowm] = se; }
    }
    __syncthreads();

    // ---- phase 3: combine 8 wave partials + target term per row -----------
    if (tid < MTILE) {
        const float o0 = sh_o0[tid];
        float M = o0;
        #pragma unroll
        for (int ww = 0; ww < 8; ++ww) M = fmaxf(M, sh_pm[ww][tid]);
        float S = __expf(2.f * (o0 - M));
        #pragma unroll
        for (int ww = 0; ww < 8; ++ww)
            S += sh_ps[ww][tid] * __expf(2.f * (sh_pm[ww][tid] - M));
        // logsumexp(2*out)/2 = M + 0.5*log(S)
        sh_row[tid] = (M + 0.5f * __logf(S)) - o0;   // = -(res0_row)
    }
    __syncthreads();

    if (tid == 0) {
        float t = 0.f;
        for (int r = 0; r < MTILE; ++r) t += sh_row[r];
        partials[blockIdx.x] = t;
    }
}

// ---------------------------------------------------------------------------
// Kernel C: deterministic fixed-order reduction of per-block partials.
// ---------------------------------------------------------------------------
__global__ __launch_bounds__(256) void reduce_kernel(const float* __restrict__ partials,
                                                     float* __restrict__ out,
                                                     int nblocks, float invN) {
    __shared__ float sh[256];
    float s = 0.f;
    for (int i = threadIdx.x; i < nblocks; i += 256) s += partials[i];
    sh[threadIdx.x] = s;
    __syncthreads();
    for (int off = 128; off >= 1; off >>= 1) {
        if (threadIdx.x < off) sh[threadIdx.x] += sh[threadIdx.x + off];
        __syncthreads();
    }
    if (threadIdx.x == 0) out[0] = sh[0] * invN;
}

// ---------------------------------------------------------------------------
extern "C" void kernel_launch(void* const* d_in, const int* in_sizes, int n_in,
                              void* d_out, int out_size, void* d_ws, size_t ws_size,
                              hipStream_t stream) {
    const float* weight       = (const float*)d_in[0];   // [V, H]
    const float* bias         = (const float*)d_in[1];   // [V]
    const float* hiddens      = (const float*)d_in[2];   // [N, H]
    const float* noise_logits = (const float*)d_in[3];   // [V]
    const int*   targets      = (const int*)d_in[4];     // [N]
    const int*   samples      = (const int*)d_in[5];     // [K]

    const int N    = in_sizes[2] / H;       // 32768
    const int nblk = N / MTILE;             // 2048

    // workspace layout (floats): Ws[K*H] | badj[K] | partials[nblk]
    float* Ws       = (float*)d_ws;
    float* badj     = Ws + KSMP * H;
    float* partials = badj + KSMP;

    gather_kernel<<<KSMP, H, 0, stream>>>(weight, bias, noise_logits, samples, Ws, badj);
    score_loss_kernel<<<nblk, 256, 0, stream>>>(weight, bias, hiddens, noise_logits,
                                                targets, Ws, badj, partials);
    reduce_kernel<<<1, 256, 0, stream>>>(partials, (float*)d_out, nblk, 1.0f / (float)N);
}